// ModelNew_36215164240264
// MI455X (gfx1250) — compile-verified
//
#include <hip/hip_runtime.h>
#include <math.h>

#define B_ 32
#define N_ 1024
#define D_ 512
#define K_ 64
#define KG_ 80   // K + G

typedef float v2f __attribute__((ext_vector_type(2)));
typedef float v8f __attribute__((ext_vector_type(8)));

__device__ __forceinline__ v8f wmma_f32x4(v2f a, v2f b, v8f c) {
    // D = A(16x4 f32) * B(4x16 f32) + C(16x16 f32)
    return __builtin_amdgcn_wmma_f32_16x16x4_f32(false, a, false, b, (short)0, c,
                                                 false, false);
}

// ---------------------------------------------------------------------------
// Kernel 1: logits = x @ clusters ; BN(eval) ; softmax over 80 ; keep 64.
// Grid: 256 blocks x 256 threads (8 waves). Wave strip = 16 rows x 80 cols.
// ---------------------------------------------------------------------------
__global__ __launch_bounds__(256) void k_assign(
    const float* __restrict__ x, const float* __restrict__ clusters,
    const float* __restrict__ bnw, const float* __restrict__ bnb,
    const float* __restrict__ bnm, const float* __restrict__ bnv,
    float* __restrict__ assign)
{
    __shared__ float ldsB[KG_ * 66];   // clusters chunk, transposed, padded
    const int tid  = threadIdx.x;
    const int wave = tid >> 5;
    const int lane = tid & 31;
    const int lo   = lane & 15;
    const int hi   = lane >> 4;
    const int rb   = blockIdx.x * 128 + wave * 16;   // first row of wave strip

    // per-lane BN scale/shift for its 5 columns
    float sc[5], sh[5];
#pragma unroll
    for (int t = 0; t < 5; ++t) {
        int c = t * 16 + lo;
        float s = bnw[c] * rsqrtf(bnv[c] + 1e-5f);
        sc[t] = s;
        sh[t] = bnb[c] - bnm[c] * s;
    }

    v8f zero = {};
    v8f acc[5];
#pragma unroll
    for (int t = 0; t < 5; ++t) acc[t] = zero;

    for (int kc = 0; kc < D_; kc += 64) {
        // stage clusters[kc..kc+63][0..79] transposed: ldsB[n*66 + kk]
        for (int e = tid; e < 64 * KG_; e += 256) {
            int kk = e / KG_, n = e - kk * KG_;
            ldsB[n * 66 + kk] = clusters[(size_t)(kc + kk) * KG_ + n];
        }
        __syncthreads();
#pragma unroll 4
        for (int kk = 0; kk < 64; kk += 4) {
            // A: x tile 16x4 (M=row, K=d); one b64 per lane
            v2f a = *(const v2f*)(x + (size_t)(rb + lo) * D_ + kc + kk + 2 * hi);
#pragma unroll
            for (int t = 0; t < 5; ++t) {
                // B: clusters^T tile 4x16 (K=d, N=col); one ds b64 per lane
                v2f b = *(const v2f*)(&ldsB[(t * 16 + lo) * 66 + kk + 2 * hi]);
                acc[t] = wmma_f32x4(a, b, acc[t]);
            }
        }
        __syncthreads();
    }

    // BN transform
#pragma unroll
    for (int t = 0; t < 5; ++t)
#pragma unroll
        for (int r = 0; r < 8; ++r) acc[t][r] = acc[t][r] * sc[t] + sh[t];

    // softmax over 80 cols; each row spans 16 lanes (one half-wave) x 5 tiles
    float mx[8], inv[8];
#pragma unroll
    for (int r = 0; r < 8; ++r) {
        float m = acc[0][r];
#pragma unroll
        for (int t = 1; t < 5; ++t) m = fmaxf(m, acc[t][r]);
#pragma unroll
        for (int msk = 1; msk <= 8; msk <<= 1)
            m = fmaxf(m, __shfl_xor(m, msk, 32));
        mx[r] = m;
    }
#pragma unroll
    for (int t = 0; t < 5; ++t)
#pragma unroll
        for (int r = 0; r < 8; ++r) acc[t][r] = __expf(acc[t][r] - mx[r]);
#pragma unroll
    for (int r = 0; r < 8; ++r) {
        float s = 0.0f;
#pragma unroll
        for (int t = 0; t < 5; ++t) s += acc[t][r];
#pragma unroll
        for (int msk = 1; msk <= 8; msk <<= 1) s += __shfl_xor(s, msk, 32);
        inv[r] = 1.0f / s;
    }
    // store first 64 columns (real clusters), coalesced per half-wave
#pragma unroll
    for (int t = 0; t < 4; ++t)
#pragma unroll
        for (int r = 0; r < 8; ++r) {
            int row = rb + r + 8 * hi;
            assign[(size_t)row * K_ + t * 16 + lo] = acc[t][r] * inv[r];
        }
}

// ---------------------------------------------------------------------------
// Kernel 2: a_sum[b,k] = sum_n assignment[b,n,k]. One block per batch.
// ---------------------------------------------------------------------------
__global__ __launch_bounds__(256) void k_asum(const float* __restrict__ assign,
                                              float* __restrict__ a_sum)
{
    __shared__ float sm[256];
    const int b = blockIdx.x;
    const int k = threadIdx.x & 63, grp = threadIdx.x >> 6;
    float s = 0.0f;
    for (int n = grp; n < N_; n += 4)
        s += assign[((size_t)b * N_ + n) * K_ + k];
    sm[threadIdx.x] = s;
    __syncthreads();
    if (threadIdx.x < 64)
        a_sum[b * K_ + threadIdx.x] = sm[threadIdx.x] + sm[threadIdx.x + 64] +
                                      sm[threadIdx.x + 128] + sm[threadIdx.x + 192];
}

// ---------------------------------------------------------------------------
// Kernel 3: vladKD[b,k,d] = sum_n a[b,n,k] * x[b,n,d].  WMMA M=k,N=d,K=n.
// Grid: 32 b x 8 d-slabs = 256 blocks; 4 waves, one 16-k tile each, 64 d.
// ---------------------------------------------------------------------------
__global__ __launch_bounds__(128) void k_vlad(
    const float* __restrict__ assign, const float* __restrict__ x,
    float* __restrict__ vladKD)
{
    __shared__ float xs[32 * 64];    // x n-chunk, shared by all 4 waves
    const int tid  = threadIdx.x;
    const int wave = tid >> 5;
    const int lane = tid & 31;
    const int lo   = lane & 15;
    const int hi   = lane >> 4;
    const int b    = blockIdx.x >> 3;
    const int d0   = (blockIdx.x & 7) * 64;

    v8f zero = {};
    v8f acc[4];
#pragma unroll
    for (int t = 0; t < 4; ++t) acc[t] = zero;

    for (int nc = 0; nc < N_; nc += 32) {
        // stage x[b, nc..nc+31, d0..d0+63] (b128 loads)
#pragma unroll
        for (int i = 0; i < 4; ++i) {
            int e = tid + i * 128;            // 512 float4 slots
            int row = e >> 4, c4 = (e & 15) * 4;
            float4 v = *(const float4*)(x + ((size_t)(b * N_ + nc + row)) * D_ + d0 + c4);
            *(float4*)(&xs[row * 64 + c4]) = v;
        }
        __syncthreads();
#pragma unroll 2
        for (int kk = 0; kk < 32; kk += 4) {
            // A: assignment^T tile 16x4 (M=k, K=n); two strided b32 per lane
            int n0 = nc + kk + 2 * hi;
            const float* pa = assign + ((size_t)(b * N_ + n0)) * K_ + wave * 16 + lo;
            v2f a;
            a.x = pa[0];
            a.y = pa[K_];
#pragma unroll
            for (int dt = 0; dt < 4; ++dt) {
                // B: x tile 4x16 (K=n, N=d) from LDS
                int nr = kk + 2 * hi, dc = dt * 16 + lo;
                v2f bv;
                bv.x = xs[nr * 64 + dc];
                bv.y = xs[(nr + 1) * 64 + dc];
                acc[dt] = wmma_f32x4(a, bv, acc[dt]);
            }
        }
        __syncthreads();
    }
    // store [b][k][d] coalesced
#pragma unroll
    for (int dt = 0; dt < 4; ++dt)
#pragma unroll
        for (int r = 0; r < 8; ++r) {
            int k = wave * 16 + r + 8 * hi;
            int d = d0 + dt * 16 + lo;
            vladKD[((size_t)b * K_ + k) * D_ + d] = acc[dt][r];
        }
}

// ---------------------------------------------------------------------------
// Kernel 4: val = vladKD - a_sum*c2 ; write unnormalized out ; accumulate ssq.
// ---------------------------------------------------------------------------
__global__ __launch_bounds__(256) void k_correct(
    const float* __restrict__ vladKD, const float* __restrict__ a_sum,
    const float* __restrict__ c2, float* __restrict__ out,
    float* __restrict__ ssq)
{
    __shared__ float red[8];
    const int b = blockIdx.x >> 7;
    const int e = ((blockIdx.x & 127) << 8) + threadIdx.x;   // 0..32767
    const int d = e >> 6, k = e & 63;
    float val = vladKD[((size_t)b * K_ + k) * D_ + d] - a_sum[b * K_ + k] * c2[e];
    out[(size_t)b * (D_ * K_) + e] = val;
    float v2 = val * val;
#pragma unroll
    for (int msk = 16; msk >= 1; msk >>= 1) v2 += __shfl_xor(v2, msk, 32);
    if ((threadIdx.x & 31) == 0) red[threadIdx.x >> 5] = v2;
    __syncthreads();
    if (threadIdx.x == 0) {
        float s = 0.0f;
#pragma unroll
        for (int i = 0; i < 8; ++i) s += red[i];
        atomicAdd(&ssq[b], s);
    }
}

// ---------------------------------------------------------------------------
// Kernel 5: double L2-normalization scale.
// v1 = v/sqrt(s+eps);  s2 = s/(s+eps);  v2 = v1/sqrt(s2+eps)
// ---------------------------------------------------------------------------
__global__ __launch_bounds__(256) void k_scale(float* __restrict__ out,
                                               const float* __restrict__ ssq)
{
    const int idx = blockIdx.x * 256 + threadIdx.x;
    const int b = idx >> 15;
    float s  = ssq[b];
    float n1 = sqrtf(s + 1e-12f);
    float s2 = s / (s + 1e-12f);
    float n2 = sqrtf(s2 + 1e-12f);
    out[idx] = out[idx] * (1.0f / (n1 * n2));
}

// ---------------------------------------------------------------------------
extern "C" void kernel_launch(void* const* d_in, const int* in_sizes, int n_in,
                              void* d_out, int out_size, void* d_ws, size_t ws_size,
                              hipStream_t stream)
{
    const float* x        = (const float*)d_in[0];
    const float* clusters = (const float*)d_in[1];
    const float* c2       = (const float*)d_in[2];   // [1,D,K] -> flat D*K
    const float* bnw      = (const float*)d_in[3];
    const float* bnb      = (const float*)d_in[4];
    const float* bnm      = (const float*)d_in[5];
    const float* bnv      = (const float*)d_in[6];
    float* out = (float*)d_out;

    char* ws = (char*)d_ws;
    float* assign = (float*)ws;                                       // 8 MB
    float* vladKD = (float*)(ws + (size_t)B_ * N_ * K_ * 4);          // 4 MB
    float* a_sum  = (float*)(ws + (size_t)B_ * N_ * K_ * 4
                                + (size_t)B_ * K_ * D_ * 4);          // 8 KB
    float* ssq    = a_sum + B_ * K_;                                  // 128 B

    hipMemsetAsync(ssq, 0, B_ * sizeof(float), stream);
    k_assign <<<dim3(256),  dim3(256), 0, stream>>>(x, clusters, bnw, bnb, bnm, bnv, assign);
    k_asum   <<<dim3(32),   dim3(256), 0, stream>>>(assign, a_sum);
    k_vlad   <<<dim3(256),  dim3(128), 0, stream>>>(assign, x, vladKD);
    k_correct<<<dim3(4096), dim3(256), 0, stream>>>(vladKD, a_sum, c2, out, ssq);
    k_scale  <<<dim3(4096), dim3(256), 0, stream>>>(out, ssq);
}